// QuantizedEmbeddingFallback_20375324852407
// MI455X (gfx1250) — compile-verified
//
#include <hip/hip_runtime.h>
#include <stdint.h>

// Quantized embedding dequant-gather for MI455X (gfx1250).
// out[tok, :] = float(qweight[x[tok], :]) * scales[x[tok], group]
// DIM=1024, 8 groups of 128. Pure bandwidth problem (~526 MB @ 23.3 TB/s).
//
// Strategy:
//  - 256 threads/block (8 wave32), each thread owns 16 B of a row (b128).
//  - 8 tokens per block, double-buffered through LDS via CDNA5 async copies
//    (global_load_async_to_lds_b128 + s_wait_asynccnt) to overlap the random
//    row gather with dequant of the previous row.
//  - Non-temporal b128 stores for the 256 MB streaming output so the 192 MB
//    L2 is reserved for qweight row reuse (~51K distinct rows of 128K vocab).

typedef __attribute__((ext_vector_type(4))) float v4f;
typedef __attribute__((ext_vector_type(4))) int   v4i;

#define QE_DIM           1024
#define QE_NGROUPS       8
#define QE_THREADS       256
#define QE_TOK_PER_BLOCK 8

__global__ __launch_bounds__(QE_THREADS) void qe_dequant_gather_kernel(
    const int*   __restrict__ x,        // [n_tokens] token ids (int32)
    const int*   __restrict__ qweight,  // [VOCAB * 1024] int codes in [-8,7]
    const float* __restrict__ scales,   // [VOCAB * 8] per-group scales
    float*       __restrict__ out,      // [n_tokens * 1024]
    int n_tokens)
{
    __shared__ __align__(16) int smem[2][QE_DIM];   // 2 x 4KB double buffer

    const int t    = threadIdx.x;                       // 0..255
    const int tok0 = blockIdx.x * QE_TOK_PER_BLOCK;
    if (tok0 >= n_tokens) return;

    // LDS byte offsets of this thread's 16B slot in each buffer.
    // (flat addr low 32 bits == wave-relative LDS offset on CDNA5)
    const unsigned lds0 = (unsigned)(uintptr_t)(&smem[0][t * 4]);
    const unsigned lds1 = (unsigned)(uintptr_t)(&smem[1][t * 4]);
    const uint64_t qw_base = (uint64_t)qweight;

    // ---- prologue: stage token 0's row into buffer 0 (async, ASYNCcnt++) ----
    {
        const int row = x[tok0];                              // uniform -> s_load
        const unsigned goff = (unsigned)row * (QE_DIM * 4u) + (unsigned)(t * 16);
        asm volatile("global_load_async_to_lds_b128 %0, %1, %2"
                     :: "v"(lds0), "v"(goff), "s"(qw_base)
                     : "memory");
    }

    #pragma unroll
    for (int i = 0; i < QE_TOK_PER_BLOCK; ++i) {
        const int tok = tok0 + i;
        if (tok >= n_tokens) break;                           // uniform per block

        // ---- issue next token's async load into the other buffer ----
        bool issued_next = false;
        if ((i + 1 < QE_TOK_PER_BLOCK) && (tok + 1 < n_tokens)) {
            const int nrow = x[tok + 1];                      // uniform -> s_load
            const unsigned goff = (unsigned)nrow * (QE_DIM * 4u) + (unsigned)(t * 16);
            const unsigned dst = ((i + 1) & 1) ? lds1 : lds0;
            asm volatile("global_load_async_to_lds_b128 %0, %1, %2"
                         :: "v"(dst), "v"(goff), "s"(qw_base)
                         : "memory");
            issued_next = true;
        }

        // ---- retire the async copy for the CURRENT buffer ----
        // async loads complete in order within a wave: if one newer load is in
        // flight, waiting for <=1 outstanding guarantees ours is done.
        if (issued_next) {
            asm volatile("s_wait_asynccnt 0x1" ::: "memory");
        } else {
            asm volatile("s_wait_asynccnt 0x0" ::: "memory");
        }

        // ---- dequantize this thread's 4 elements ----
        const int row = x[tok];
        // elements [4t .. 4t+3] never straddle a 128-wide group; group = t/32,
        // i.e. one scale per wave32 -> single broadcast load per wave.
        const float s = scales[row * QE_NGROUPS + (t >> 5)];

        const v4i q = *reinterpret_cast<const v4i*>(&smem[i & 1][t * 4]);  // ds_load_b128
        v4f r;
        r.x = (float)q.x * s;
        r.y = (float)q.y * s;
        r.z = (float)q.z * s;
        r.w = (float)q.w * s;

        // streaming, write-once output: non-temporal b128 store keeps L2 free
        // for qweight row reuse.
        v4f* dst = reinterpret_cast<v4f*>(out + (size_t)tok * QE_DIM + t * 4);
        __builtin_nontemporal_store(r, dst);
    }
}

extern "C" void kernel_launch(void* const* d_in, const int* in_sizes, int n_in,
                              void* d_out, int out_size, void* d_ws, size_t ws_size,
                              hipStream_t stream) {
    const int*   x       = (const int*)d_in[0];     // [32*2048] indices
    const int*   qweight = (const int*)d_in[1];     // [128000*1024] int codes
    const float* scales  = (const float*)d_in[2];   // [128000*8]
    float*       out     = (float*)d_out;           // [32*2048*1024]

    const int n_tokens = in_sizes[0];               // 65536
    const int blocks = (n_tokens + QE_TOK_PER_BLOCK - 1) / QE_TOK_PER_BLOCK;

    qe_dequant_gather_kernel<<<blocks, QE_THREADS, 0, stream>>>(
        x, qweight, scales, out, n_tokens);
}